// RiemannianConv_55800215110045
// MI455X (gfx1250) — compile-verified
//
#include <hip/hip_runtime.h>

typedef __attribute__((ext_vector_type(2))) float v2f;
typedef __attribute__((ext_vector_type(8))) float v8f;

#define N_NODES 100000
#define N_EDGES 1600000
#define D 64
#define NODE_TILES (N_NODES / 16)   // 6250, exact

// ---------------------------------------------------------------------------
// Kernel A: zero the scatter accumulators in workspace
// ---------------------------------------------------------------------------
__global__ __launch_bounds__(256)
void rc_zero_kernel(float* __restrict__ agg, float* __restrict__ deg) {
    int i = blockIdx.x * blockDim.x + threadIdx.x;
    if (i < N_NODES * D) agg[i] = 0.0f;
    if (i < N_NODES)     deg[i] = 0.0f;
}

// ---------------------------------------------------------------------------
// Kernel B: h = x @ W^T + b via V_WMMA_F32_16X16X4_F32 (fp32-exact path)
// One wave32 per (16-node tile, 16-output-col tile).
//   A (16x4, f32): lanes 0-15 -> M=lane, K={0,1}; lanes 16-31 -> K={2,3}
//   B (4x16, f32): lanes 0-15 -> N=lane, K={0,1}; lanes 16-31 -> K={2,3}
//   C/D (16x16):   VGPR r -> M = r + 8*(lane>>4), N = lane&15
// ---------------------------------------------------------------------------
__global__ __launch_bounds__(256)
void rc_gemm_wmma_kernel(const float* __restrict__ x,
                         const float* __restrict__ W,
                         const float* __restrict__ b,
                         float* __restrict__ h) {
    const int lane  = threadIdx.x & 31;
    const int wave  = blockIdx.x * 8 + (threadIdx.x >> 5);
    const int nodeTile = wave >> 2;          // 0 .. 6249
    const int colTile  = wave & 3;           // 0 .. 3  (4 x 16 = 64 output cols)
    if (nodeTile >= NODE_TILES) return;      // never taken (uniform), EXEC stays all-1

    const int m0 = nodeTile * 16;
    const int n0 = colTile * 16;
    const int lo = lane & 15;                // M (for A) / N (for B,C,D)
    const int hi = lane >> 4;                // selects K pair {0,1} vs {2,3}

    const float* __restrict__ xrow = x + (size_t)(m0 + lo) * D;  // A row
    const float* __restrict__ wrow = W + (size_t)(n0 + lo) * D;  // B col = W row

    v8f c = {};
    #pragma unroll
    for (int kk = 0; kk < D; kk += 4) {
        const int k = kk + 2 * hi;           // even -> 8B-aligned float2 loads
        v2f a = *(const v2f*)(xrow + k);
        v2f bv = *(const v2f*)(wrow + k);
        // (neg_a, A, neg_b, B, c_mod, C, reuse_a, reuse_b)
        c = __builtin_amdgcn_wmma_f32_16x16x4_f32(false, a, false, bv,
                                                  (short)0, c, false, false);
    }

    const float bias = b[n0 + lo];
    #pragma unroll
    for (int r = 0; r < 8; ++r) {
        const int m = m0 + r + 8 * hi;
        h[(size_t)m * D + n0 + lo] = c[r] + bias;
    }
}

// ---------------------------------------------------------------------------
// Kernel C: edge gather + scatter-add. 16 lanes per edge, float4 per lane.
// All of h/agg fits in the 192 MB L2, so random gathers + atomics stay on-chip.
// ---------------------------------------------------------------------------
__global__ __launch_bounds__(256)
void rc_edge_kernel(const long long* __restrict__ edge_index,  // [2, E] int64
                    const float* __restrict__ edge_metric,     // [E]
                    const float* __restrict__ h,               // [N, 64]
                    float* __restrict__ agg,                   // [N, 64]
                    float* __restrict__ deg) {                 // [N]
    const int t = blockIdx.x * blockDim.x + threadIdx.x;
    const int e = t >> 4;                    // 16 lanes per edge
    if (e >= N_EDGES) return;
    const int sub = t & 15;

    const int row = (int)edge_index[e];               // destination node
    const int col = (int)edge_index[N_EDGES + e];     // source node
    const float m = edge_metric[e];

    const float4 hv = *(const float4*)(h + (size_t)col * D + sub * 4);
    float* __restrict__ dst = agg + (size_t)row * D + sub * 4;
    atomicAdd(dst + 0, hv.x * m);
    atomicAdd(dst + 1, hv.y * m);
    atomicAdd(dst + 2, hv.z * m);
    atomicAdd(dst + 3, hv.w * m);
    if (sub == 0) atomicAdd(deg + row, 1.0f);
}

// ---------------------------------------------------------------------------
// Kernel D: out = h + scale * agg / max(deg, 1)   (in place: h lives in d_out)
// ---------------------------------------------------------------------------
__global__ __launch_bounds__(256)
void rc_finalize_kernel(const float* __restrict__ agg,
                        const float* __restrict__ deg,
                        const float* __restrict__ scale,
                        float* __restrict__ out) {
    const int i = blockIdx.x * blockDim.x + threadIdx.x;
    if (i >= N_NODES * D) return;
    const int n = i >> 6;
    float d = deg[n];
    d = (d < 1.0f) ? 1.0f : d;
    out[i] = out[i] + scale[0] * (agg[i] / d);
}

// ---------------------------------------------------------------------------
// Host-side launcher (graph-capture safe: only kernel launches on `stream`)
// ---------------------------------------------------------------------------
extern "C" void kernel_launch(void* const* d_in, const int* in_sizes, int n_in,
                              void* d_out, int out_size, void* d_ws, size_t ws_size,
                              hipStream_t stream) {
    const float*     x     = (const float*)d_in[0];      // [N, 64]
    const float*     W     = (const float*)d_in[1];      // [64, 64]
    const float*     b     = (const float*)d_in[2];      // [64]
    const float*     scale = (const float*)d_in[3];      // [1]
    const long long* ei    = (const long long*)d_in[4];  // [2, E] int64
    const float*     em    = (const float*)d_in[5];      // [E]

    float* out = (float*)d_out;            // holds h, then the final result
    float* agg = (float*)d_ws;             // [N*64] floats
    float* deg = agg + (size_t)N_NODES * D; // [N] floats

    // A: zero accumulators (covers N*64 elements; deg covered within range)
    rc_zero_kernel<<<(N_NODES * D + 255) / 256, 256, 0, stream>>>(agg, deg);

    // B: WMMA GEMM -> h in d_out. 6250 node tiles * 4 col tiles = 25000 waves
    rc_gemm_wmma_kernel<<<(NODE_TILES * 4 + 7) / 8, 256, 0, stream>>>(x, W, b, out);

    // C: edge gather/scatter. 16 threads per edge -> E*16 threads
    rc_edge_kernel<<<(N_EDGES * 16) / 256, 256, 0, stream>>>(ei, em, out, agg, deg);

    // D: finalize in place
    rc_finalize_kernel<<<(N_NODES * D + 255) / 256, 256, 0, stream>>>(agg, deg, scale, out);
}